// SemiCRF_23510650978502
// MI455X (gfx1250) — compile-verified
//
#include <hip/hip_runtime.h>

#define B_ 4
#define T_ 512
#define L_ 96
#define W_ 63
#define NEGINF (-1e30f)

typedef unsigned int du4 __attribute__((ext_vector_type(4)));
typedef int          di8 __attribute__((ext_vector_type(8)));
typedef int          di4 __attribute__((ext_vector_type(4)));
typedef float        v8f __attribute__((ext_vector_type(8)));
typedef float        v2f __attribute__((ext_vector_type(2)));

// ---------------------------------------------------------------------------
// Kernel 0: precompute E[j][lp][ln] = exp((j+1) * trans[lp][ln]); t-invariant.
// 63*96*96 floats = 2.32 MB, lives in d_ws, streamed from L2 every step.
// ---------------------------------------------------------------------------
__global__ void build_E_kernel(const float* __restrict__ trans,
                               float* __restrict__ E) {
  int i = blockIdx.x * blockDim.x + threadIdx.x;
  if (i < W_ * L_ * L_) {
    int j = i / (L_ * L_);
    int r = i - j * (L_ * L_);
    E[i] = expf((float)(j + 1) * trans[r]);
  }
}

// ---------------------------------------------------------------------------
// TDM: async 2D tiled load (rows x 96 floats, row stride T_*L_ elements)
// from global into LDS. Descriptor per CDNA5 ISA.
// ---------------------------------------------------------------------------
__device__ __forceinline__ void tdm_load_seg(unsigned int lds_off,
                                             const float* gptr, int rows) {
  unsigned long long ga = (unsigned long long)(const void*)gptr;
  du4 g0;
  g0.x = 1u;                                    // count=1, no gather, no restore
  g0.y = lds_off;                               // LDS byte address
  g0.z = (unsigned int)ga;                      // global_addr[31:0]
  g0.w = (unsigned int)(ga >> 32) | (2u << 30); // global_addr[56:32] | type=2
  di8 g1;
  g1[0] = (2 << 16);                 // workgroup_mask=0, data_size=2 (4B)
  g1[1] = (L_ << 16);                // tensor_dim0[15:0]=96 in bits[31:16]
  g1[2] = ((rows & 0xFFFF) << 16);   // tensor_dim0 hi=0; tensor_dim1 lo16=rows
  g1[3] = (L_ << 16);                // tensor_dim1 hi=0; tile_dim0=96
  g1[4] = (rows & 0xFFFF);           // tile_dim1=rows; tile_dim2=0
  g1[5] = T_ * L_;                   // tensor_dim0_stride lo32 (elements)
  g1[6] = 0;                         // stride hi16 / dim1_stride lo (unused, 2D)
  g1[7] = 0;
  di4 z4 = {0, 0, 0, 0};
#if defined(__clang_major__) && (__clang_major__ >= 23)
  di8 z8 = {0, 0, 0, 0, 0, 0, 0, 0};
  __builtin_amdgcn_tensor_load_to_lds(g0, g1, z4, z4, z8, 0);
#else
  __builtin_amdgcn_tensor_load_to_lds(g0, g1, z4, z4, 0);
#endif
}

// ---------------------------------------------------------------------------
// Kernel 1: persistent scan. One workgroup (1024 threads = 32 waves) per batch.
// All recurrence state lives in LDS; only workgroup barriers.
// ---------------------------------------------------------------------------
__global__ void __launch_bounds__(1024, 1)
semicrf_scan(const float* __restrict__ seg, const float* __restrict__ trans,
             const float* __restrict__ E, float* __restrict__ out) {
  extern __shared__ char smem[];
  float* s_seg  = (float*)smem;                       // 2 * 63*96  (TDM dbl buf)
  float* s_Ahat = (float*)(smem + 48384);             // 63*96 ring: exp(alpha-rowmax)
  float* s_rmax = (float*)(smem + 48384 + 24192);     // 63 ring (pad to 256B)
  float* s_tq   = (float*)(smem + 48384 + 24192 + 256); // 64*96: tvals then Q (in place)
  float* s_mcol = s_tq + 64 * L_;                     // 96
  float* s_cs   = s_mcol + L_;                        // 96 (WMMA column sums)
  float* s_new  = s_cs + L_;                          // 96
  float* s_z    = s_new + L_;                         // 2*96 (z-row dbl buf)

  const int tid  = threadIdx.x;
  const int lane = tid & 31;
  const int wave = tid >> 5;
  const int b    = blockIdx.x;
  const float* segb = seg + (size_t)b * T_ * T_ * L_;
  const unsigned int lds_seg_off = (unsigned int)(size_t)s_seg;

  float bosv = 0.f;
  if (tid < L_) bosv = trans[L_ * L_ + tid];

  // alpha row 0 = exp(seg[b,0,0,:] + bos)  (exp-space value, per reference quirk)
  if (tid < L_) s_new[tid] = expf(segb[tid] + bosv);
  __syncthreads();
  if (tid < 32) {
    float v = fmaxf(fmaxf(s_new[tid], s_new[tid + 32]), s_new[tid + 64]);
    for (int o = 16; o > 0; o >>= 1) v = fmaxf(v, __shfl_xor(v, o, 32));
    if (lane == 0) s_rmax[0] = v;
  }
  __syncthreads();
  if (tid < L_) s_Ahat[tid] = expf(s_new[tid] - s_rmax[0]);
  // z row for t=1: seg[b,0,1,:]
  if (tid >= 128 && tid < 128 + L_)
    s_z[L_ + (tid - 128)] = segb[(size_t)1 * L_ + (tid - 128)];
  // TDM prologue: window for t=1 (s0=1, 1 row) into buffer 1
  if (wave == 0)
    tdm_load_seg(lds_seg_off + 24192u, segb + ((size_t)1 * T_ + 1) * L_, 1);

  for (int t = 1; t < T_; ++t) {
    const int cur  = t & 1;
    const int base = (t > W_) ? (t - W_) : 0;
    const int m    = (t < W_) ? t : W_;

    if (wave == 0) __builtin_amdgcn_s_wait_tensorcnt(0); // window t landed
    __syncthreads();                                     // + prev row published

    // prefetch seg window for t+1 into the other buffer (overlaps compute)
    if (wave == 0 && t + 1 < T_) {
      const int tn = t + 1;
      const int basen = (tn > W_) ? (tn - W_) : 0;
      const int mn = (tn < W_) ? tn : W_;
      tdm_load_seg(lds_seg_off + (unsigned)(tn & 1) * 24192u,
                   segb + ((size_t)(basen + 1) * T_ + tn) * L_, mn);
    }
    // prefetch next z row (needed while t+1 <= W)
    float zpref = 0.f;
    const bool zp = (t < W_) && (tid >= 128) && (tid < 128 + L_);
    if (zp) zpref = segb[(size_t)(t + 1) * L_ + (tid - 128)];

    // ---- G stage: tq[j][ln] = rowmax + w_j*seg + log(sum_lp Ahat*E) ----
    const float* segcur = s_seg + cur * (W_ * L_);
    for (int task = wave; task < 3 * m; task += 32) {
      const int j   = task / 3;
      const int lnb = task - j * 3;
      const int ln  = lnb * 32 + lane;
      const int row = (base + j) % W_;
      const float* Ah = s_Ahat + row * L_;
      const float* Ep = E + (size_t)j * (L_ * L_) + ln;
      float acc = 0.f;
#pragma unroll 8
      for (int lp = 0; lp < L_; ++lp)
        acc = fmaf(Ah[lp], Ep[(size_t)lp * L_], acc);   // acc >= 1 always
      s_tq[j * L_ + ln] =
          s_rmax[row] + (float)(j + 1) * segcur[j * L_ + ln] + logf(acc);
    }
    if (zp) s_z[((t + 1) & 1) * L_ + (tid - 128)] = zpref;
    __syncthreads();

    // ---- per-column max over valid j (+ boundary z term as row 63) ----
    if (tid < L_) {
      float mc = NEGINF;
      for (int j = 0; j < m; ++j) mc = fmaxf(mc, s_tq[j * L_ + tid]);
      if (t <= W_) {
        float zv = (s_z[cur * L_ + tid] + bosv) * (float)(t + 1);
        s_tq[W_ * L_ + tid] = zv;
        mc = fmaxf(mc, zv);
      }
      s_mcol[tid] = mc;
    }
    __syncthreads();

    // ---- Q = exp(tq - mcol) in place; invalid rows -> 0 ----
    for (int e = tid; e < 64 * L_; e += 1024) {
      const int j  = e / L_;
      const int ln = e - j * L_;
      const bool valid = (j < m) | ((j == W_) & (t <= W_));
      s_tq[e] = valid ? expf(s_tq[e] - s_mcol[ln]) : 0.f;
    }
    __syncthreads();

    // ---- column sums of Q (64x96) via V_WMMA_F32_16X16X4_F32, A = ones ----
    // C[m,n] = sum_k 1 * B[k,n]; any K-lane striping of B sums correctly.
    if (wave < 6) {
      const int n0   = wave * 16;
      const int lanN = lane & 15;
      const int half = lane >> 4;
      v8f c = {0.f, 0.f, 0.f, 0.f, 0.f, 0.f, 0.f, 0.f};
      v2f a = {1.f, 1.f};
#pragma unroll
      for (int k = 0; k < 16; ++k) {
        v2f bv;
        bv.x = s_tq[(k * 4 + half) * L_ + n0 + lanN];
        bv.y = s_tq[(k * 4 + 2 + half) * L_ + n0 + lanN];
        c = __builtin_amdgcn_wmma_f32_16x16x4_f32(false, a, false, bv,
                                                  (short)0, c, false, false);
      }
      if (lane < 16) s_cs[n0 + lanN] = c[0];  // row M=0 = column sums
    }
    __syncthreads();

    if (tid < L_) s_new[tid] = s_mcol[tid] + logf(s_cs[tid]);
    __syncthreads();

    const int slot = t % W_;
    if (tid < 32) {
      float v = fmaxf(fmaxf(s_new[tid], s_new[tid + 32]), s_new[tid + 64]);
      for (int o = 16; o > 0; o >>= 1) v = fmaxf(v, __shfl_xor(v, o, 32));
      if (lane == 0) s_rmax[slot] = v;
    }
    __syncthreads();
    if (tid < L_) s_Ahat[slot * L_ + tid] = expf(s_new[tid] - s_rmax[slot]);
    // loop-top barrier publishes the new row
  }

  if (tid == 0) {
    const int slot = (T_ - 1) % W_;
    float s = 0.f;
    for (int ln = 0; ln < L_; ++ln) s += s_Ahat[slot * L_ + ln];
    out[b] = s_rmax[slot] + logf(s);
  }
}

// ---------------------------------------------------------------------------
extern "C" void kernel_launch(void* const* d_in, const int* in_sizes, int n_in,
                              void* d_out, int out_size, void* d_ws,
                              size_t ws_size, hipStream_t stream) {
  const float* seg   = (const float*)d_in[0];  // (B,T,T,L) f32
  const float* trans = (const float*)d_in[1];  // (L+1,L)   f32
  float* E = (float*)d_ws;                     // 63*96*96 f32 = 2.32 MB

  const int ne = W_ * L_ * L_;
  build_E_kernel<<<(ne + 255) / 256, 256, 0, stream>>>(trans, E);

  const size_t smem = 48384 + 24192 + 256 + (64 * L_) * 4 + 3 * L_ * 4 + 2 * L_ * 4;
  semicrf_scan<<<B_, 1024, smem, stream>>>(seg, trans, E, (float*)d_out);
}